// AtomTypeLinear_44848048505186
// MI455X (gfx1250) — compile-verified
//
#include <hip/hip_runtime.h>

// Per-atom typed linear layer: out[n,:] = x[n,:] @ W[type[n]] + b[type[n]]
// Strategy: bucket rows by type on-device so 16-row tiles share one weight
// matrix, then use full-precision FP32 WMMA (v_wmma_f32_16x16x4_f32) tiles.

typedef __attribute__((ext_vector_type(2))) float v2f;
typedef __attribute__((ext_vector_type(8))) float v8f;

#define IN_DIM 64
#define OUT_DIM 64
#define LDS_STRIDE 68   // 16B-aligned rows, conflict-free column reads

// ---------------- bucketing kernels ----------------

__global__ void atl_zero_counts(int* counts, int T) {
    int i = blockIdx.x * blockDim.x + threadIdx.x;
    if (i < T) counts[i] = 0;
}

__global__ void atl_hist(const int* __restrict__ types, int n,
                         int* __restrict__ counts) {
    int i = blockIdx.x * blockDim.x + threadIdx.x;
    if (i < n) atomicAdd(&counts[types[i]], 1);
}

// Exclusive scan of 16-padded per-type counts (T=64: trivial scalar loop).
__global__ void atl_prefix(const int* __restrict__ counts, int T,
                           int* __restrict__ offsets,
                           int* __restrict__ cursors,
                           int* __restrict__ ntiles) {
    if (threadIdx.x == 0 && blockIdx.x == 0) {
        int off = 0;
        for (int t = 0; t < T; ++t) {
            offsets[t] = off;
            cursors[t] = off;
            off += ((counts[t] + 15) >> 4) << 4;   // pad segment to 16 rows
        }
        offsets[T] = off;
        ntiles[0] = off >> 4;
    }
}

__global__ void atl_scatter(const int* __restrict__ types, int n,
                            int* __restrict__ cursors,
                            int* __restrict__ sorted) {
    int i = blockIdx.x * blockDim.x + threadIdx.x;
    if (i < n) {
        int pos = atomicAdd(&cursors[types[i]], 1);
        sorted[pos] = i;
    }
}

// ---------------- WMMA GEMM kernel ----------------
// One block = one 16-row tile (single type). 4 wave32s; wave w owns output
// columns [16w, 16w+16). Each wave: 16x v_wmma_f32_16x16x4_f32 over K=64,
// accumulator pre-seeded with the per-type bias (column-only dependence).

__global__ __launch_bounds__(128)
void atl_wmma_kernel(const float* __restrict__ x,
                     const float* __restrict__ mat,
                     const float* __restrict__ bias,
                     float* __restrict__ out,
                     const int* __restrict__ counts,
                     const int* __restrict__ offsets,
                     const int* __restrict__ ntiles,
                     const int* __restrict__ sorted) {
    __shared__ float ldsA[16 * LDS_STRIDE];
    __shared__ int s_meta[3];   // type, segment start, segment count

    if ((int)blockIdx.x >= ntiles[0]) return;   // block-uniform early exit
    const int tid = threadIdx.x;
    const int tile16 = (int)blockIdx.x << 4;

    if (tid == 0) {
        int t = 0;
        while (offsets[t + 1] <= tile16) ++t;   // 64-entry scan, SGPR-friendly
        s_meta[0] = t;
        s_meta[1] = offsets[t];
        s_meta[2] = counts[t];
    }
    __syncthreads();
    const int t     = s_meta[0];
    const int local = tile16 - s_meta[1];       // tile's offset within segment
    const int cnt   = s_meta[2];

    // Gather 16 x-rows into LDS (zero padding rows). 128 threads x 2 float4.
    for (int q = 0; q < 2; ++q) {
        int slot = tid + (q << 7);              // 0..255
        int r  = slot >> 4;                     // row 0..15
        int c4 = (slot & 15) << 2;              // float col (x4)
        float4 v = make_float4(0.f, 0.f, 0.f, 0.f);
        if (local + r < cnt) {
            int idx = sorted[tile16 + r];
            v = *(const float4*)(x + (size_t)idx * IN_DIM + c4);
        }
        *(float4*)(&ldsA[r * LDS_STRIDE + c4]) = v;
    }
    __syncthreads();

    const int lane = tid & 31;
    const int wave = tid >> 5;
    const int nb   = wave << 4;     // output-column base for this wave
    const int half = lane >> 4;     // 0: lanes 0-15, 1: lanes 16-31
    const int n    = lane & 15;     // N index (== M index for A fragment)

    // Seed accumulator with bias: every element this lane owns has column nb+n.
    const float bv = bias[t * OUT_DIM + nb + n];
    v8f acc = {bv, bv, bv, bv, bv, bv, bv, bv};

    const float* __restrict__ Wt   = mat + (size_t)t * (IN_DIM * OUT_DIM);
    const float* __restrict__ Arow = &ldsA[n * LDS_STRIDE];

    #pragma unroll
    for (int kt = 0; kt < 16; ++kt) {
        const int kk = (kt << 2) + (half << 1); // ISA 16x4 A layout: half-wave K split
        v2f a, b;
        a.x = Arow[kk];                         // A[m][kk],   m = lane&15
        a.y = Arow[kk + 1];                     // A[m][kk+1]
        b.x = Wt[kk * OUT_DIM + nb + n];        // B[kk][nb+n]   (L2-resident)
        b.y = Wt[(kk + 1) * OUT_DIM + nb + n];  // B[kk+1][nb+n]
        acc = __builtin_amdgcn_wmma_f32_16x16x4_f32(
                  /*neg_a=*/false, a, /*neg_b=*/false, b,
                  /*c_mod=*/(short)0, acc, /*reuse_a=*/false, /*reuse_b=*/false);
    }

    // C/D layout: VGPR j holds row (j + 8*half), column nb+n.
    #pragma unroll
    for (int j = 0; j < 8; ++j) {
        const int row = j + (half << 3);
        if (local + row < cnt) {
            const int idx = sorted[tile16 + row];
            out[(size_t)idx * OUT_DIM + nb + n] = acc[j];
        }
    }
}

// ---------------- host-side launcher ----------------

extern "C" void kernel_launch(void* const* d_in, const int* in_sizes, int n_in,
                              void* d_out, int out_size, void* d_ws, size_t ws_size,
                              hipStream_t stream) {
    const float* x    = (const float*)d_in[0];
    const int*   ty   = (const int*)d_in[1];
    const float* mat  = (const float*)d_in[2];
    const float* bias = (const float*)d_in[3];
    float*       out  = (float*)d_out;

    const int N = in_sizes[1];                          // one type per row
    const int T = in_sizes[2] / (IN_DIM * OUT_DIM);     // number of types

    // Workspace layout (ints): counts[T] | offsets[T+1] | cursors[T] |
    //                          ntiles[1] | sorted[N + 16T]
    int* ws      = (int*)d_ws;
    int* counts  = ws;
    int* offsets = counts + T;
    int* cursors = offsets + (T + 1);
    int* ntiles  = cursors + T;
    int* sorted  = ntiles + 1;

    atl_zero_counts<<<(T + 63) / 64, 64, 0, stream>>>(counts, T);
    atl_hist<<<(N + 255) / 256, 256, 0, stream>>>(ty, N, counts);
    atl_prefix<<<1, 1, 0, stream>>>(counts, T, offsets, cursors, ntiles);
    atl_scatter<<<(N + 255) / 256, 256, 0, stream>>>(ty, N, cursors, sorted);

    const int max_tiles = (N + 15) / 16 + T;            // upper bound; blocks early-exit
    atl_wmma_kernel<<<max_tiles, 128, 0, stream>>>(x, mat, bias, out,
                                                   counts, offsets, ntiles, sorted);
}